// ROI_Layer_43198781063657
// MI455X (gfx1250) — compile-verified
//
#include <hip/hip_runtime.h>
#include <stdint.h>

// ---------------- problem constants (match reference) ----------------
#define BATCH 8
#define NANCH 262144
#define PRE_NMS 2000
#define PROP 1000
#define NMS_THR 0.7f

// ---------------- tuning ----------------
#define NBUCKET 32768          // 15-bit key histogram
#define KEY_SHIFT 17
#define CAND_CAP 8192          // candidate cap per batch (expected ~2000-4100)
#define CHUNK_PAIRS 4096       // score pairs per TDM tile (= 8192 f32 = 32 KB)
#define BLOCKS_PER_BATCH 8
#define CHUNKS_PER_BLOCK ((NANCH / CHUNK_PAIRS) / BLOCKS_PER_BATCH)  // 8
#define JBLOCKS ((PRE_NMS + 31) / 32)   // 63

typedef unsigned int u32;
typedef unsigned long long u64;
typedef u32 v4u __attribute__((ext_vector_type(4)));
typedef int  v8i __attribute__((ext_vector_type(8)));
typedef int  v4i __attribute__((ext_vector_type(4)));

// ---------------- workspace layout (bytes) ----------------
#define HIST_OFF  0
#define CANDN_OFF ((size_t)BATCH * NBUCKET * 4)          // 1 MiB
#define THR_OFF   (CANDN_OFF + 64)
#define ZERO_BYTES (CANDN_OFF + 512)                     // memset hist + counters
#define CAND_OFF  (CANDN_OFF + 512)
#define BOXES_OFF (CAND_OFF + (size_t)BATCH * CAND_CAP * 8)
#define MASK_OFF  (BOXES_OFF + (size_t)BATCH * PRE_NMS * 16)

// Monotone (descending-friendly) key for IEEE f32.
__device__ __forceinline__ u32 fkey(float f) {
  u32 u = __float_as_uint(f);
  return (u & 0x80000000u) ? ~u : (u | 0x80000000u);
}

// ---------------- Tensor Data Mover: 1-D f32 tile -> LDS ----------------
// D# per CDNA5 ISA ch.8: group0 {count=1, lds_addr, global_addr, type=2},
// group1 {data_size=4B, tensor_dim0=n, tensor_dim1=1, tile_dim0=n, stride0=n}.
// amdgpu-toolchain (clang-23) builtin arity: 6 args
//   (uint32x4 g0, int32x8 g1, int32x4 g2, int32x4 g3, int32x8 extra, i32 cpol)
__device__ __forceinline__ void tdm_load_f32_1d(u32 lds_byte_off,
                                                const float* __restrict__ gsrc,
                                                u32 nelem) {
#if __has_builtin(__builtin_amdgcn_tensor_load_to_lds)
  uint64_t ga = (uint64_t)(uintptr_t)gsrc;
  v4u g0;
  g0.x = 1u;                                              // count=1, user mode
  g0.y = lds_byte_off;                                    // lds_addr
  g0.z = (u32)ga;                                         // global_addr[31:0]
  g0.w = (u32)((ga >> 32) & 0x01FFFFFFull) | (2u << 30);  // addr[56:32] | type=2
  v8i g1;
  g1[0] = 0x00020000;                    // data_size=2 (4 bytes), no multicast
  g1[1] = (int)(nelem << 16);            // tensor_dim0[15:0] @ [31:16]
  g1[2] = (int)((nelem >> 16) | (1u << 16)); // tensor_dim0[31:16] | tensor_dim1=1
  g1[3] = (int)(nelem << 16);            // tile_dim0 @ [31:16]
  g1[4] = 0;                             // tile_dim1=0, tile_dim2=0 (1-D tile)
  g1[5] = (int)nelem;                    // tensor_dim0_stride[31:0]
  g1[6] = 0;
  g1[7] = 0;
  v4i gz  = {0, 0, 0, 0};
  v8i gz8 = {0, 0, 0, 0, 0, 0, 0, 0};
  __builtin_amdgcn_tensor_load_to_lds(g0, g1, gz, gz, gz8, 0);
#else
  (void)lds_byte_off; (void)gsrc; (void)nelem;
#endif
}

__device__ __forceinline__ void tdm_wait(int pending) {
#if __has_builtin(__builtin_amdgcn_s_wait_tensorcnt)
  if (pending) __builtin_amdgcn_s_wait_tensorcnt(1);
  else         __builtin_amdgcn_s_wait_tensorcnt(0);
#else
  (void)pending;
#endif
}

// ---------------- stage 1: key histogram over fg scores -----------------
// Single-wave blocks (TDM issue is per-wave and EXEC-independent, so one wave
// per block guarantees exactly one DMA per call). Double-buffered TDM stream.
__global__ __launch_bounds__(32) void k_hist(const float* __restrict__ scores,
                                             u32* __restrict__ hist) {
  __shared__ float buf[2][CHUNK_PAIRS * 2];  // 2 x 32 KB
  const int blk  = blockIdx.x;               // 0..63
  const int b    = blk / BLOCKS_PER_BATCH;
  const int sub  = blk % BLOCKS_PER_BATCH;
  const int lane = threadIdx.x;
  const float* base = scores + (size_t)b * NANCH * 2;
  const int chunk0 = sub * CHUNKS_PER_BLOCK;
  u32* __restrict__ h = hist + (size_t)b * NBUCKET;

#if __has_builtin(__builtin_amdgcn_tensor_load_to_lds)
  tdm_load_f32_1d((u32)(uintptr_t)&buf[0][0],
                  base + (size_t)chunk0 * CHUNK_PAIRS * 2, CHUNK_PAIRS * 2);
  for (int c = 0; c < CHUNKS_PER_BLOCK; ++c) {
    const int more = (c + 1 < CHUNKS_PER_BLOCK);
    if (more)
      tdm_load_f32_1d((u32)(uintptr_t)&buf[(c + 1) & 1][0],
                      base + (size_t)(chunk0 + c + 1) * CHUNK_PAIRS * 2,
                      CHUNK_PAIRS * 2);
    tdm_wait(more);                          // in-order TDM: buf[c&1] is ready
    __asm__ volatile("" ::: "memory");       // LDS written by DMA: force reload
    const float* lb = &buf[c & 1][0];
    for (int p = lane; p < CHUNK_PAIRS; p += 32) {
      u32 key = fkey(lb[2 * p + 1]);         // fg channel
      atomicAdd(&h[key >> KEY_SHIFT], 1u);
    }
  }
#else
  // Fallback: straight global reads (no TDM builtin on this toolchain).
  for (int c = 0; c < CHUNKS_PER_BLOCK; ++c) {
    const float* gb = base + (size_t)(chunk0 + c) * CHUNK_PAIRS * 2;
    for (int p = lane; p < CHUNK_PAIRS; p += 32) {
      u32 key = fkey(gb[2 * p + 1]);
      atomicAdd(&h[key >> KEY_SHIFT], 1u);
    }
  }
#endif
}

// ---------------- stage 2: per-batch threshold bucket --------------------
__global__ void k_thresh(const u32* __restrict__ hist, u32* __restrict__ thr) {
  const int b = blockIdx.x;
  if (threadIdx.x != 0) return;
  const u32* h = hist + (size_t)b * NBUCKET;
  u32 acc = 0, t = 0;
  for (int bk = NBUCKET - 1; bk >= 0; --bk) {
    acc += h[bk];
    if (acc >= PRE_NMS) { t = (u32)bk; break; }
  }
  thr[b] = t;
}

// ---------------- stage 3: compact candidates ----------------------------
__global__ __launch_bounds__(256) void k_compact(const float* __restrict__ scores,
                                                 const u32* __restrict__ thr,
                                                 u32* __restrict__ candN,
                                                 u64* __restrict__ cand) {
  const size_t g = (size_t)blockIdx.x * 256 + threadIdx.x;  // b*N + i
  if (g >= (size_t)BATCH * NANCH) return;
  __builtin_prefetch(scores + 2 * g + 4096, 0, 0);          // global_prefetch_b8
  const int b = (int)(g / NANCH);
  const u32 i = (u32)(g % NANCH);
  const u32 key = fkey(scores[2 * g + 1]);
  if ((key >> KEY_SHIFT) >= thr[b]) {
    u32 pos = atomicAdd(&candN[b], 1u);
    if (pos < CAND_CAP)
      cand[(size_t)b * CAND_CAP + pos] = ((u64)key << 32) | (u32)(~i);
  }
}

// ---------------- stage 4: bitonic sort + box decode ---------------------
__global__ __launch_bounds__(1024) void k_sort_boxes(const u64* __restrict__ cand,
                                                     const u32* __restrict__ candN,
                                                     const float* __restrict__ deltas,
                                                     const float* __restrict__ anchors,
                                                     float* __restrict__ boxes) {
  __shared__ u64 s[CAND_CAP];                 // 64 KB
  const int b = blockIdx.x, tid = threadIdx.x;
  u32 C = candN[b]; if (C > CAND_CAP) C = CAND_CAP;
  for (int i = tid; i < CAND_CAP; i += 1024)
    s[i] = (i < (int)C) ? cand[(size_t)b * CAND_CAP + i] : 0ull;
  __syncthreads();
  for (int k = 2; k <= CAND_CAP; k <<= 1) {
    for (int j = k >> 1; j > 0; j >>= 1) {
      for (int i = tid; i < CAND_CAP; i += 1024) {
        const int ixj = i ^ j;
        if (ixj > i) {
          u64 a = s[i], c2 = s[ixj];
          const bool desc = ((i & k) == 0);   // overall descending sort
          if (desc ? (a < c2) : (a > c2)) { s[i] = c2; s[ixj] = a; }
        }
      }
      __syncthreads();
    }
  }
  for (int r = tid; r < PRE_NMS; r += 1024) {
    const u32 idx = ~(u32)(s[r] & 0xFFFFFFFFull);
    float* o = boxes + ((size_t)b * PRE_NMS + r) * 4;
    if (idx >= NANCH) { o[0] = o[1] = o[2] = o[3] = 0.f; continue; }
    const size_t off = ((size_t)b * NANCH + idx) * 4;
    const float d0 = deltas[off + 0] * 0.1f, d1 = deltas[off + 1] * 0.1f;
    const float d2 = deltas[off + 2] * 0.2f, d3 = deltas[off + 3] * 0.2f;
    const float a0 = anchors[off + 0], a1 = anchors[off + 1];
    const float a2 = anchors[off + 2], a3 = anchors[off + 3];
    float h = a2 - a0, w = a3 - a1;
    const float cy = a0 + 0.5f * h + d0 * h;
    const float cx = a1 + 0.5f * w + d1 * w;
    h *= expf(d2); w *= expf(d3);
    float y1 = cy - 0.5f * h, x1 = cx - 0.5f * w;
    float y2 = cy + 0.5f * h, x2 = cx + 0.5f * w;
    o[0] = fminf(fmaxf(y1, 0.f), 1.f);
    o[1] = fminf(fmaxf(x1, 0.f), 1.f);
    o[2] = fminf(fmaxf(y2, 0.f), 1.f);
    o[3] = fminf(fmaxf(x2, 0.f), 1.f);
  }
}

// ---------------- stage 5: IoU suppression bitmask -----------------------
__global__ __launch_bounds__(256) void k_mask(const float* __restrict__ boxes,
                                              u32* __restrict__ mask) {
  const int g = blockIdx.x * 256 + threadIdx.x;
  const int total = BATCH * PRE_NMS * JBLOCKS;
  if (g >= total) return;
  const int jb = g % JBLOCKS;
  const int i  = (g / JBLOCKS) % PRE_NMS;
  const int b  = g / (JBLOCKS * PRE_NMS);
  const float* bb = boxes + (size_t)b * PRE_NMS * 4;
  const float y1 = bb[i * 4 + 0], x1 = bb[i * 4 + 1];
  const float y2 = bb[i * 4 + 2], x2 = bb[i * 4 + 3];
  const float ai = (y2 - y1) * (x2 - x1);
  u32 m = 0;
  const int j0 = jb * 32;
  for (int t = 0; t < 32; ++t) {
    const int j = j0 + t;
    if (j >= PRE_NMS) break;
    const float by1 = bb[j * 4 + 0], bx1 = bb[j * 4 + 1];
    const float by2 = bb[j * 4 + 2], bx2 = bb[j * 4 + 3];
    const float aj = (by2 - by1) * (bx2 - bx1);
    const float iy = fmaxf(fminf(y2, by2) - fmaxf(y1, by1), 0.f);
    const float ix = fmaxf(fminf(x2, bx2) - fmaxf(x1, bx1), 0.f);
    const float inter = iy * ix;
    const float iou = inter / fmaxf(ai + aj - inter, 1e-10f);
    if (iou > NMS_THR) m |= (1u << t);
  }
  mask[((size_t)b * PRE_NMS + i) * JBLOCKS + jb] = m;
}

// ---------------- stage 6: greedy-NMS reduce + output --------------------
__global__ __launch_bounds__(64) void k_nms_out(const float* __restrict__ boxes,
                                                const u32* __restrict__ mask,
                                                float* __restrict__ out) {
  __shared__ u32 rem[JBLOCKS];
  const int b = blockIdx.x, tid = threadIdx.x;
  for (int i = tid; i < PROP * 4; i += 64) out[(size_t)b * PROP * 4 + i] = 0.f;
  for (int w = tid; w < JBLOCKS; w += 64) rem[w] = 0u;
  __syncthreads();
  int rank = 0;
  for (int i = 0; i < PRE_NMS; ++i) {
    const bool kept = ((rem[i >> 5] >> (i & 31)) & 1u) == 0u;  // uniform
    __syncthreads();                         // all reads before any writes
    if (kept) {
      if (rank < PROP && tid < 4)
        out[((size_t)b * PROP + rank) * 4 + tid] =
            boxes[((size_t)b * PRE_NMS + i) * 4 + tid];
      ++rank;
      const u32* mrow = mask + ((size_t)b * PRE_NMS + i) * JBLOCKS;
      for (int w = tid; w < JBLOCKS; w += 64) rem[w] |= mrow[w];
    }
    __syncthreads();                         // writes visible for next read
  }
}

// ---------------- host launcher ------------------------------------------
extern "C" void kernel_launch(void* const* d_in, const int* in_sizes, int n_in,
                              void* d_out, int out_size, void* d_ws, size_t ws_size,
                              hipStream_t stream) {
  const float* scores  = (const float*)d_in[0];  // (B,N,2)
  const float* deltas  = (const float*)d_in[1];  // (B,N,4)
  const float* anchors = (const float*)d_in[2];  // (B,N,4)
  float* out = (float*)d_out;                    // (B,1000,4)
  char* ws = (char*)d_ws;

  u32* hist   = (u32*)(ws + HIST_OFF);
  u32* candN  = (u32*)(ws + CANDN_OFF);
  u32* thr    = (u32*)(ws + THR_OFF);
  u64* cand   = (u64*)(ws + CAND_OFF);
  float* boxes = (float*)(ws + BOXES_OFF);
  u32* mask   = (u32*)(ws + MASK_OFF);

  (void)hipMemsetAsync(ws, 0, ZERO_BYTES, stream);  // hist + counters + thresholds

  k_hist<<<BATCH * BLOCKS_PER_BATCH, 32, 0, stream>>>(scores, hist);
  k_thresh<<<BATCH, 32, 0, stream>>>(hist, thr);
  const size_t nElem = (size_t)BATCH * NANCH;
  k_compact<<<(unsigned)((nElem + 255) / 256), 256, 0, stream>>>(scores, thr, candN, cand);
  k_sort_boxes<<<BATCH, 1024, 0, stream>>>(cand, candN, deltas, anchors, boxes);
  const int totalMask = BATCH * PRE_NMS * JBLOCKS;
  k_mask<<<(totalMask + 255) / 256, 256, 0, stream>>>(boxes, mask);
  k_nms_out<<<BATCH, 64, 0, stream>>>(boxes, mask, out);
}